// BilinearPooling_17609365914058
// MI455X (gfx1250) — compile-verified
//
#include <hip/hip_runtime.h>
#include <math.h>

// BilinearPooling + row L2-normalize, MI455X (gfx1250, wave32).
//
// Roofline: 192 MB total traffic, ~84 MFLOP -> AI ~0.44 FLOP/B -> pure
// bandwidth problem (~8.2 us floor at 23.3 TB/s). WMMA cannot help; the
// win is one-pass streaming with b128 accesses, NT hints (working set ~= L2),
// and CDNA5 async global->LDS DMA for the visual stream overlapped with
// audio register loads.

typedef __attribute__((ext_vector_type(4))) float v4f;

#define EPS 1e-12f
#define D 2048
#define BLOCK 256   // 8 wave32s; each thread owns 2 float4 chunks of the row

__global__ __launch_bounds__(BLOCK) void bilinear_l2norm_kernel(
    const float* __restrict__ audio,
    const float* __restrict__ visual,
    float* __restrict__ out)
{
    const int row  = blockIdx.x;
    const int t    = threadIdx.x;      // 0..255
    const int wave = t >> 5;
    const int lane = t & 31;

    __shared__ __align__(16) float vis_lds[D];   // 8 KB staging for visual row
    __shared__ float red_s[BLOCK / 32];          // cross-wave partial: sum(visual)
    __shared__ float red_a[BLOCK / 32];          // cross-wave partial: sum(audio^2)

    const float* __restrict__ arow = audio  + (size_t)row * D;
    const float* __restrict__ vrow = visual + (size_t)row * D;
    float*       __restrict__ orow = out    + (size_t)row * D;

    // ---- Async DMA: visual row -> LDS (two b128 per thread, coalesced) ----
    // GVS addressing: SADDR(64b SGPR pair) + VADDR(32b byte offset).
    // LDS dest address taken from VGPR (low 32 bits of the flat LDS pointer).
    {
        unsigned ldsa0 = (unsigned)(size_t)(&vis_lds[4 * t]);
        unsigned ldsa1 = (unsigned)(size_t)(&vis_lds[1024 + 4 * t]);
        unsigned goff0 = 16u * (unsigned)t;            // bytes, chunk 0
        unsigned goff1 = 4096u + 16u * (unsigned)t;    // bytes, chunk 1
        asm volatile("global_load_async_to_lds_b128 %0, %1, %2 offset:0"
                     :: "v"(ldsa0), "v"(goff0), "s"(vrow) : "memory");
        asm volatile("global_load_async_to_lds_b128 %0, %1, %2 offset:0"
                     :: "v"(ldsa1), "v"(goff1), "s"(vrow) : "memory");
    }

    // ---- Meanwhile: audio chunks -> registers (streamed, non-temporal) ----
    v4f a0 = __builtin_nontemporal_load((const v4f*)(arow + 4 * t));
    v4f a1 = __builtin_nontemporal_load((const v4f*)(arow + 1024 + 4 * t));

    float sa = a0.x * a0.x + a0.y * a0.y + a0.z * a0.z + a0.w * a0.w
             + a1.x * a1.x + a1.y * a1.y + a1.z * a1.z + a1.w * a1.w;

    // ---- Wait for async DMA, read own LDS region, partial visual sum ----
    asm volatile("s_wait_asynccnt 0x0" ::: "memory");
    v4f v0 = *(const v4f*)(&vis_lds[4 * t]);
    v4f v1 = *(const v4f*)(&vis_lds[1024 + 4 * t]);
    float sv = v0.x + v0.y + v0.z + v0.w + v1.x + v1.y + v1.z + v1.w;

    // ---- wave32 butterfly reduction (warpSize == 32 on gfx1250) ----
    #pragma unroll
    for (int m = 16; m > 0; m >>= 1) {
        sv += __shfl_xor(sv, m, 32);
        sa += __shfl_xor(sa, m, 32);
    }
    if (lane == 0) {
        red_s[wave] = sv;
        red_a[wave] = sa;
    }
    __syncthreads();

    float S = 0.0f, A = 0.0f;
    #pragma unroll
    for (int w = 0; w < BLOCK / 32; ++w) {
        S += red_s[w];
        A += red_a[w];
    }

    // pooled = audio * S ; ||pooled|| = sqrt(S^2 * sum(audio^2))
    float norm  = sqrtf(S * S * A);
    float scale = S / fmaxf(norm, EPS);

    // ---- Streamed non-temporal b128 stores ----
    v4f o0 = a0 * scale;
    v4f o1 = a1 * scale;
    __builtin_nontemporal_store(o0, (v4f*)(orow + 4 * t));
    __builtin_nontemporal_store(o1, (v4f*)(orow + 1024 + 4 * t));
}

extern "C" void kernel_launch(void* const* d_in, const int* in_sizes, int n_in,
                              void* d_out, int out_size, void* d_ws, size_t ws_size,
                              hipStream_t stream)
{
    const float* audio  = (const float*)d_in[0];
    const float* visual = (const float*)d_in[1];
    float* out = (float*)d_out;

    const int B = in_sizes[0] / D;   // 8192 for the reference shapes

    bilinear_l2norm_kernel<<<B, BLOCK, 0, stream>>>(audio, visual, out);
}